// LinearCNNLayer_39410619908201
// MI455X (gfx1250) — compile-verified
//
#include <hip/hip_runtime.h>

// CDNA5 (gfx1250) implicit-GEMM 3x3 valid conv via V_WMMA_F32_16X16X4_F32.
// Reference "sparse linear CNN" == dense conv2d(x, w, valid) + bias.

typedef __attribute__((ext_vector_type(2))) float v2f;
typedef __attribute__((ext_vector_type(8))) float v8f;

#define SI   32
#define KS   3
#define IC   16
#define OC   32
#define SO   30            // SI - KS + 1
#define NB   32            // batch
#define KDIM (IC * KS * KS)   // 144 = GEMM K
#define NTOT (NB * SO * SO)   // 28800 = GEMM N (divisible by 16 -> EXEC all-1s)
#define XCH  (SI * SI)        // 1024 floats per channel
#define XB   (IC * SI * SI)   // 16384 floats per batch
#define OUTB (OC * SO * SO)   // 28800 floats per batch
#define WAVES_PER_BLOCK 4

__global__ __launch_bounds__(128) void
conv3x3_wmma_f32(const float* __restrict__ x,
                 const float* __restrict__ w,
                 const float* __restrict__ bias,
                 float* __restrict__ out)
{
    // Weights [OC][KDIM] staged once per block (18 KB), plus im2col offset table.
    __shared__ float sW[OC * KDIM];
    __shared__ int   sOff[KDIM];

    const int tid = threadIdx.x;
    for (int t = tid; t < OC * KDIM; t += 128) sW[t] = w[t];
    if (tid < KDIM) {
        const int ic = tid / 9, r = tid % 9;
        sOff[tid] = ic * XCH + (r / 3) * SI + (r % 3);
    }
    __syncthreads();

    const int wave  = tid >> 5;
    const int lane  = tid & 31;
    const int laneN = lane & 15;       // GEMM column within the 16-wide N tile
    const bool hiHalf = lane >= 16;    // lanes 16-31 carry K=2,3 (A) / K=2,3 rows (B)

    // One wave -> one 32(M) x 16(N) output tile; two f32 accumulators share B.
    const int ntile = blockIdx.x * WAVES_PER_BLOCK + wave;
    const int n = ntile * 16 + laneN;          // flattened (b, i, j)
    const int b = n / (SO * SO);
    const int p = n % (SO * SO);
    const int i = p / SO;
    const int j = p % SO;
    const float* xb = x + b * XB + i * SI + j; // top-left of this lane's patch

    v8f c0 = {};   // output rows y = 0..15
    v8f c1 = {};   // output rows y = 16..31
    const int koff = hiHalf ? 2 : 0;

    #pragma unroll 4
    for (int kk = 0; kk < KDIM / 4; ++kk) {
        const int kb = 4 * kk + koff;

        // A fragments (16x4 f32): lane m=laneN(+16), float2 = K pair from LDS.
        const v2f a0 = *(const v2f*)&sW[laneN * KDIM + kb];
        const v2f a1 = *(const v2f*)&sW[(laneN + 16) * KDIM + kb];

        // B fragment (4x16 f32): column n = laneN, rows kb, kb+1 (im2col gather).
        v2f bf;
        bf.x = xb[sOff[kb]];
        bf.y = xb[sOff[kb + 1]];

        // D = A*B + C  (full fp32, round-to-nearest-even)
        c0 = __builtin_amdgcn_wmma_f32_16x16x4_f32(false, a0, false, bf,
                                                   (short)0, c0, false, false);
        c1 = __builtin_amdgcn_wmma_f32_16x16x4_f32(false, a1, false, bf,
                                                   (short)0, c1, false, false);
    }

    // C/D layout: VGPR r -> M=r (lanes 0-15) or M=r+8 (lanes 16-31), N=laneN.
    const int mbase = hiHalf ? 8 : 0;
    float* ob = out + b * OUTB + p;
    #pragma unroll
    for (int r = 0; r < 8; ++r) {
        const int y0 = mbase + r;
        ob[y0 * (SO * SO)]        = c0[r] + bias[y0];
        ob[(y0 + 16) * (SO * SO)] = c1[r] + bias[y0 + 16];
    }
}

extern "C" void kernel_launch(void* const* d_in, const int* in_sizes, int n_in,
                              void* d_out, int out_size, void* d_ws, size_t ws_size,
                              hipStream_t stream) {
    (void)in_sizes; (void)n_in; (void)d_ws; (void)ws_size; (void)out_size;
    const float* x    = (const float*)d_in[0];   // [B, IC, SI, SI]
    const float* w    = (const float*)d_in[1];   // [OC*IC*K*K] == [OC][144] row-major
    const float* bias = (const float*)d_in[2];   // [OC]
    // d_in[3..5] = rows/cols/param_idx: regenerated analytically, unused.
    float* out = (float*)d_out;                  // [B, OC, SO, SO]

    const int ntiles = NTOT / 16;                         // 1800
    const int blocks = ntiles / WAVES_PER_BLOCK;          // 450, exact
    conv3x3_wmma_f32<<<blocks, 128, 0, stream>>>(x, w, bias, out);
}